// LogSumExpWirelength_68298569941404
// MI455X (gfx1250) — compile-verified
//
#include <hip/hip_runtime.h>

// Log-sum-exp wirelength, atomic formulation.
//   ws layout: per net n, 4 consecutive f32: {sum exp(+x/g), sum exp(+y/g),
//                                             sum exp(-x/g), sum exp(-y/g)}
//   (16B per net -> one cacheline locality for the 4 atomics of a pin,
//    32MB total -> resident in MI455X's 192MB L2 where the atomic units live)

typedef float v4f __attribute__((ext_vector_type(4)));

#define LSE_LOG2E 1.44269504088896340736f
#define LSE_LN2   0.69314718055994530942f

__global__ void lse_wl_zero_kernel(float* __restrict__ ws,
                                   float* __restrict__ out,
                                   int num_nets) {
    int i = blockIdx.x * blockDim.x + threadIdx.x;
    if (i < num_nets) {
        v4f z = {0.f, 0.f, 0.f, 0.f};
        *(v4f*)(ws + 4ull * (unsigned)i) = z;
    }
    if (i == 0) out[0] = 0.f;
}

__global__ void lse_wl_pin_kernel(const float* __restrict__ pos,
                                  const int* __restrict__ pin2net,
                                  const float* __restrict__ gamma_p,
                                  float* __restrict__ ws,
                                  int num_pins) {
    int i = blockIdx.x * blockDim.x + threadIdx.x;
    if (i >= num_pins) return;

    // scalar (uniform) gamma; fold 1/gamma * log2(e) into one multiplier
    float c = LSE_LOG2E / gamma_p[0];

    // streamed exactly once -> non-temporal so the 32MB accumulator
    // array stays resident in L2
    float x  = __builtin_nontemporal_load(pos + i);
    float y  = __builtin_nontemporal_load(pos + num_pins + i);
    int  net = __builtin_nontemporal_load(pin2net + i);

    float sx = x * c;
    float sy = y * c;
    float epx = __builtin_amdgcn_exp2f(sx);    // exp(+x/g)
    float epy = __builtin_amdgcn_exp2f(sy);
    float enx = __builtin_amdgcn_exp2f(-sx);   // exp(-x/g)
    float eny = __builtin_amdgcn_exp2f(-sy);

    float* base = ws + 4ull * (unsigned)net;
    // hardware f32 atomic add at L2 (no CAS loop)
    unsafeAtomicAdd(base + 0, epx);
    unsafeAtomicAdd(base + 1, epy);
    unsafeAtomicAdd(base + 2, enx);
    unsafeAtomicAdd(base + 3, eny);
}

__global__ void lse_wl_reduce_kernel(const float* __restrict__ ws,
                                     const unsigned char* __restrict__ net_mask,
                                     const float* __restrict__ gamma_p,
                                     float* __restrict__ out,
                                     int num_nets) {
    float acc = 0.f;   // accumulates log2 terms; scaled by gamma*ln2 at the end
    int stride = gridDim.x * blockDim.x;
    for (int i = blockIdx.x * blockDim.x + threadIdx.x; i < num_nets; i += stride) {
        v4f v = __builtin_nontemporal_load((const v4f*)ws + i);
        if (net_mask[i]) {
            float a = v.x > 0.f ? v.x : 1.f;   // empty-net guard: log(1)=0
            float b = v.y > 0.f ? v.y : 1.f;
            float d = v.z > 0.f ? v.z : 1.f;
            float e = v.w > 0.f ? v.w : 1.f;
            acc += __builtin_amdgcn_logf(a) + __builtin_amdgcn_logf(b)
                 + __builtin_amdgcn_logf(d) + __builtin_amdgcn_logf(e);
        }
    }

    // wave32 butterfly reduction
    #pragma unroll
    for (int off = 16; off > 0; off >>= 1)
        acc += __shfl_xor(acc, off, 32);

    __shared__ float lds[32];
    int lane = threadIdx.x & 31;
    int wave = threadIdx.x >> 5;
    if (lane == 0) lds[wave] = acc;
    __syncthreads();

    if (wave == 0) {
        int nwaves = blockDim.x >> 5;
        float v = (lane < nwaves) ? lds[lane] : 0.f;
        #pragma unroll
        for (int off = 16; off > 0; off >>= 1)
            v += __shfl_xor(v, off, 32);
        if (lane == 0) {
            float g = gamma_p[0];
            unsafeAtomicAdd(out, v * g * LSE_LN2);
        }
    }
}

extern "C" void kernel_launch(void* const* d_in, const int* in_sizes, int n_in,
                              void* d_out, int out_size, void* d_ws, size_t ws_size,
                              hipStream_t stream) {
    const float*         pos     = (const float*)d_in[0];
    const int*           pin2net = (const int*)d_in[1];
    const unsigned char* mask    = (const unsigned char*)d_in[2];
    const float*         gamma   = (const float*)d_in[3];
    float* out = (float*)d_out;
    float* ws  = (float*)d_ws;    // needs num_nets * 4 * sizeof(float) = 32 MB

    const int num_pins = in_sizes[0] / 2;   // pos = [x pins | y pins]
    const int num_nets = in_sizes[2];

    const int TB = 256;  // 8 wave32 per block

    lse_wl_zero_kernel<<<(num_nets + TB - 1) / TB, TB, 0, stream>>>(ws, out, num_nets);
    lse_wl_pin_kernel<<<(num_pins + TB - 1) / TB, TB, 0, stream>>>(pos, pin2net, gamma, ws, num_pins);
    lse_wl_reduce_kernel<<<1024, TB, 0, stream>>>(ws, mask, gamma, out, num_nets);
}